// MyConv_7258494730825
// MI455X (gfx1250) — compile-verified
//
#include <hip/hip_runtime.h>
#include <hip/hip_bf16.h>

typedef __attribute__((ext_vector_type(2))) float v2f;
typedef __attribute__((ext_vector_type(8))) float v8f;

#define GN 20000
#define GE 300000
#define ND 256
#define ED 64
#define BN_EPS 1e-5f

// ---------------------------------------------------------------------------
// Kernel 1: h = x ; zero first 256 floats after h in ws (BN sums only; 256-thread blocks)
// ---------------------------------------------------------------------------
__global__ void gine_init(const float* __restrict__ x, float* __restrict__ h,
                          float* __restrict__ bn_acc) {
    long i = (long)blockIdx.x * blockDim.x + threadIdx.x;
    if (i < (long)GN * ND) h[i] = x[i];
    if (blockIdx.x == 0 && threadIdx.x < 2 * ND) bn_acc[threadIdx.x] = 0.0f;
}

// ---------------------------------------------------------------------------
// Kernel 2: edge messages via f32 WMMA + fused ReLU + scatter-add
//   grid: ceil(E/128) blocks of 512 threads (16 waves)
//   wave w owns output columns [16w, 16w+16); B frags (We) register-resident
// ---------------------------------------------------------------------------
__global__ void __launch_bounds__(512)
gine_edge(const float* __restrict__ x, const int* __restrict__ ei,
          const float* __restrict__ ea, const float* __restrict__ We,
          const float* __restrict__ be, float* __restrict__ h) {
    __shared__ float lA[16 * 68];                 // 16x64 tile, stride 68 (pad)

    const int tid   = threadIdx.x;
    const int wave  = tid >> 5;
    const int lane  = tid & 31;
    const int r16   = lane & 15;                  // row (A) / col (B,C) in tile
    const int kb    = (lane >> 4) << 1;           // 0 or 2 (K sub-pair)
    const int ncol  = wave * 16 + r16;            // global output column

    // B fragments: We[k][ncol] for k = 4s+kb, 4s+kb+1  (32 VGPRs, loaded once)
    v2f bfrag[16];
#pragma unroll
    for (int s = 0; s < 16; ++s) {
        int k = 4 * s + kb;
        bfrag[s].x = We[k * ND + ncol];
        bfrag[s].y = We[(k + 1) * ND + ncol];
    }
    const float bias = be[ncol];

    const long ebase = (long)blockIdx.x * 128;
    for (int m = 0; m < 8; ++m) {
        const long e0 = ebase + (long)m * 16;
        __syncthreads();                          // protect prior-iter LDS reads
        {   // stage edge_attr[e0..e0+16) x 64 -> LDS (2 floats per thread)
            int idx = tid * 2;
            int r = idx >> 6, c = idx & 63;
            long e = e0 + r; if (e >= GE) e = GE - 1;   // clamp tail (loads only)
            v2f v = *(const v2f*)(ea + e * ED + c);
            lA[r * 68 + c]     = v.x;
            lA[r * 68 + c + 1] = v.y;
        }
        __syncthreads();

        v8f acc = {};
#pragma unroll
        for (int s = 0; s < 16; ++s) {
            v2f a = *(const v2f*)&lA[r16 * 68 + 4 * s + kb];
            acc = __builtin_amdgcn_wmma_f32_16x16x4_f32(
                false, a, false, bfrag[s], (short)0, acc, false, false);
        }

        // epilogue: msg = relu(wmma + x[src] + be); atomic scatter into h[dst]
#pragma unroll
        for (int r = 0; r < 8; ++r) {
            int mrow = r + ((lane >> 4) << 3);    // C/D layout: M = r (+8 hi half)
            long e = e0 + mrow;
            if (e < GE) {
                int s = ei[e];                    // src row
                int d = ei[GE + e];               // dst row
                float v = acc[r] + x[(long)s * ND + ncol] + bias;
                v = v > 0.0f ? v : 0.0f;
                atomicAdd(&h[(long)d * ND + ncol], v);
            }
        }
    }
}

// ---------------------------------------------------------------------------
// Kernel 3: out = h @ W + b   (pre-BN), f32 WMMA, K=256 (64 steps)
//   grid: 250 blocks x 512 threads; 5 row-tiles (16 rows) per block
//   W fragments register-resident across all 5 tiles (128 VGPRs)
// ---------------------------------------------------------------------------
__global__ void __launch_bounds__(512)
gine_node(const float* __restrict__ h, const float* __restrict__ W,
          const float* __restrict__ b, float* __restrict__ out) {
    __shared__ float lA[16 * 260];                // 16x256 tile, stride 260

    const int tid  = threadIdx.x;
    const int wave = tid >> 5;
    const int lane = tid & 31;
    const int r16  = lane & 15;
    const int kb   = (lane >> 4) << 1;
    const int ncol = wave * 16 + r16;

    v2f bfrag[64];
#pragma unroll
    for (int s = 0; s < 64; ++s) {
        int k = 4 * s + kb;
        bfrag[s].x = W[k * ND + ncol];
        bfrag[s].y = W[(k + 1) * ND + ncol];
    }
    const float bias = b[ncol];

    for (int rb = 0; rb < 5; ++rb) {
        const int rowblk = (blockIdx.x * 5 + rb) * 16;   // 20000 = 250*5*16
        __syncthreads();
        {   // stage h[rowblk..+16) x 256 -> LDS (8 floats per thread)
            int r = tid >> 5;
            int c = (tid & 31) * 8;
            const float4* src = (const float4*)(h + (long)(rowblk + r) * ND + c);
            float4 v0 = src[0], v1 = src[1];
            float* dp = &lA[r * 260 + c];
            *(float4*)dp = v0;
            *(float4*)(dp + 4) = v1;
        }
        __syncthreads();

        v8f acc = {};
#pragma unroll
        for (int s = 0; s < 64; ++s) {
            v2f a = *(const v2f*)&lA[r16 * 260 + 4 * s + kb];
            acc = __builtin_amdgcn_wmma_f32_16x16x4_f32(
                false, a, false, bfrag[s], (short)0, acc, false, false);
        }
#pragma unroll
        for (int r = 0; r < 8; ++r) {
            int mrow = r + ((lane >> 4) << 3);
            out[(long)(rowblk + mrow) * ND + ncol] = acc[r] + bias;
        }
    }
}

// ---------------------------------------------------------------------------
// Kernel 4: BN column stats (sum, sumsq) — 200 blocks x 256 threads, 100 rows ea
// ---------------------------------------------------------------------------
__global__ void gine_bnstats(const float* __restrict__ out, float* __restrict__ acc) {
    const int c = threadIdx.x;                    // column 0..255
    const long r0 = (long)blockIdx.x * 100;
    float s = 0.0f, sq = 0.0f;
    for (int r = 0; r < 100; ++r) {
        float v = out[(r0 + r) * ND + c];
        s += v; sq += v * v;
    }
    atomicAdd(&acc[c], s);
    atomicAdd(&acc[ND + c], sq);
}

// ---------------------------------------------------------------------------
// Kernel 5: BN finalize -> scale/shift (1 block x 256 threads)
// ---------------------------------------------------------------------------
__global__ void gine_bnfin(const float* __restrict__ acc,
                           const float* __restrict__ gamma,
                           const float* __restrict__ beta,
                           float* __restrict__ ss) {
    const int c = threadIdx.x;
    const float inv_n = 1.0f / (float)GN;
    float mean = acc[c] * inv_n;
    float var  = acc[ND + c] * inv_n - mean * mean;
    float inv  = rsqrtf(var + BN_EPS);
    float g    = gamma[c] * inv;
    ss[c]      = g;                               // scale
    ss[ND + c] = beta[c] - mean * g;              // shift
}

// ---------------------------------------------------------------------------
// Kernel 6: apply BN + ReLU in place
// ---------------------------------------------------------------------------
__global__ void gine_bnapply(float* __restrict__ out, const float* __restrict__ ss) {
    long i = (long)blockIdx.x * blockDim.x + threadIdx.x;
    if (i < (long)GN * ND) {
        int c = (int)(i & (ND - 1));
        float v = out[i] * ss[c] + ss[ND + c];
        out[i] = v > 0.0f ? v : 0.0f;
    }
}

// ---------------------------------------------------------------------------
extern "C" void kernel_launch(void* const* d_in, const int* in_sizes, int n_in,
                              void* d_out, int out_size, void* d_ws, size_t ws_size,
                              hipStream_t stream) {
    const float* x     = (const float*)d_in[0];
    const int*   ei    = (const int*)d_in[1];
    const float* ea    = (const float*)d_in[2];
    const float* We    = (const float*)d_in[3];
    const float* be    = (const float*)d_in[4];
    const float* W     = (const float*)d_in[5];
    const float* b     = (const float*)d_in[6];
    const float* gamma = (const float*)d_in[7];
    const float* beta  = (const float*)d_in[8];
    float* out = (float*)d_out;

    float* h      = (float*)d_ws;                 // [GN*ND]
    float* bn_acc = h + (size_t)GN * ND;          // [2*ND] sum | sumsq
    float* ss     = bn_acc + 2 * ND;              // [2*ND] scale | shift

    const long total = (long)GN * ND;             // 5,120,000
    gine_init<<<(int)((total + 255) / 256), 256, 0, stream>>>(x, h, bn_acc);
    gine_edge<<<(GE + 127) / 128, 512, 0, stream>>>(x, ei, ea, We, be, h);
    gine_node<<<250, 512, 0, stream>>>(h, W, b, out);
    gine_bnstats<<<200, 256, 0, stream>>>(out, bn_acc);
    gine_bnfin<<<1, 256, 0, stream>>>(bn_acc, gamma, beta, ss);
    gine_bnapply<<<(int)((total + 255) / 256), 256, 0, stream>>>(out, ss);
}